// CrossAttentionMatrix_43808666419284
// MI455X (gfx1250) — compile-verified
//
#include <hip/hip_runtime.h>

typedef float v2f __attribute__((ext_vector_type(2)));
typedef float v8f __attribute__((ext_vector_type(8)));
typedef float f4  __attribute__((ext_vector_type(4)));

#define HW   65536   // 256*256 spatial positions
#define CCH  64      // channels
#define NB   16      // batch
#define SUBS 4       // phase-1 blocks per (b,c) plane
#define PLANE_CHUNK (HW / SUBS)   // 16384 floats per block

// ---------------------------------------------------------------------------
// Phase 1a: partial sums of y over each (b,c) plane.
// 4096 blocks (4 per plane) x 256 threads; each thread sums 64 floats via
// 16 nontemporal float4 loads (coalesced 4KB per iteration per block).
// Writes one partial per block -> no atomics, no ws zero-init needed.
// ---------------------------------------------------------------------------
__global__ __launch_bounds__(256) void reduce_y_partial(
    const float* __restrict__ y, float* __restrict__ partial) {
  const int plane = blockIdx.x >> 2;
  const int sub   = blockIdx.x & 3;
  const float* p  = y + (size_t)plane * HW + (size_t)sub * PLANE_CHUNK;
  const int tid   = threadIdx.x;

  float sum = 0.0f;
#pragma unroll
  for (int j = 0; j < PLANE_CHUNK / (256 * 4); ++j) {
    // streaming read-once data: non-temporal to keep L2 for phase-2 v reuse
    f4 d = __builtin_nontemporal_load((const f4*)(p + j * 1024 + tid * 4));
    sum += d.x + d.y + d.z + d.w;
  }

  __shared__ float s[256];
  s[tid] = sum;
  __syncthreads();
#pragma unroll
  for (int off = 128; off > 0; off >>= 1) {
    if (tid < off) s[tid] += s[tid + off];
    __syncthreads();
  }
  if (tid == 0) partial[plane * SUBS + sub] = s[0];
}

// ---------------------------------------------------------------------------
// Phase 1b: fold SUBS partials per plane and apply the full scale
//   v[b,c] = (sum_p y[b,c,p]) / (hw * hw * c)   (== mean/hw/c), 2^-38 exact.
// ---------------------------------------------------------------------------
__global__ __launch_bounds__(1024) void finalize_v(
    const float* __restrict__ partial, float* __restrict__ v) {
  const int p = threadIdx.x;  // 0..1023 = (b,c) plane id
  float s = partial[p * SUBS + 0] + partial[p * SUBS + 1] +
            partial[p * SUBS + 2] + partial[p * SUBS + 3];
  const float scale = 1.0f / ((float)HW * (float)HW * (float)CCH);  // 2^-38
  v[p] = s * scale;
}

// ---------------------------------------------------------------------------
// Phase 2: G[b,i] = sum_c x[b,c,i] * v[b,c]  via V_WMMA_F32_16X16X4_F32.
//
// Per wave (wave32): one 16-position output tile, K = 64 channels in 16
// chunks of 4. A (16x4 f32) holds v broadcast down M:
//   lanes 0-15:  VGPR0=K0, VGPR1=K1 ; lanes 16-31: VGPR0=K2, VGPR1=K3
// B (4x16 f32) holds the x tile, N = lane%16:
//   VGPR0 = rows K0 (lanes 0-15) / K2 (lanes 16-31), VGPR1 = K1 / K3
// D[m,n] = sum_k v_k * x[k,n] (independent of m); lanes 0-15 store acc[0].
// Every x element is loaded exactly once (coalesced b32, NT hint).
// ---------------------------------------------------------------------------
__global__ __launch_bounds__(256) void gram_wmma(
    const float* __restrict__ x, const float* __restrict__ v,
    float* __restrict__ out) {
  const int tid   = threadIdx.x;
  const int wave  = tid >> 5;
  const int lane  = tid & 31;
  const int t     = blockIdx.x * 8 + wave;   // tile id, 65536 total
  const int b     = t >> 12;                 // 4096 tiles per batch
  const int ibase = (t & 4095) << 4;         // 16 spatial positions per tile

  const float* xb = x + (size_t)b * CCH * HW;
  const float* vb = v + b * CCH;
  const int n     = lane & 15;   // N index within tile
  const int khalf = lane >> 4;   // which pair of K rows this half-wave holds

  v8f acc = {};
#pragma unroll
  for (int c4 = 0; c4 < CCH / 4; ++c4) {
    const int c0 = (c4 << 2) + (khalf << 1);
    v2f A, B;
    A.x = vb[c0];
    A.y = vb[c0 + 1];
    B.x = __builtin_nontemporal_load(xb + (size_t)c0 * HW + ibase + n);
    B.y = __builtin_nontemporal_load(xb + (size_t)(c0 + 1) * HW + ibase + n);
    acc = __builtin_amdgcn_wmma_f32_16x16x4_f32(
        /*neg_a=*/false, A, /*neg_b=*/false, B,
        /*c_mod=*/(short)0, acc, /*reuse_a=*/false, /*reuse_b=*/false);
  }

  if (lane < 16) out[(size_t)b * HW + ibase + n] = acc[0];
}

// ---------------------------------------------------------------------------
extern "C" void kernel_launch(void* const* d_in, const int* in_sizes, int n_in,
                              void* d_out, int out_size, void* d_ws,
                              size_t ws_size, hipStream_t stream) {
  const float* x = (const float*)d_in[0];
  const float* y = (const float*)d_in[1];
  float* out = (float*)d_out;

  // Workspace layout: [ partials: 1024*SUBS floats | v: 1024 floats ] = 20 KB
  float* partial = (float*)d_ws;
  float* v       = partial + NB * CCH * SUBS;

  reduce_y_partial<<<NB * CCH * SUBS, 256, 0, stream>>>(y, partial);
  finalize_v<<<1, NB * CCH, 0, stream>>>(partial, v);
  gram_wmma<<<(NB * HW / 16) / 8, 256, 0, stream>>>(x, v, out);
}